// MessageAggregator_19499151524027
// MI455X (gfx1250) — compile-verified
//
#include <hip/hip_runtime.h>
#include <hip/hip_bf16.h>

#define DM 64

// ---- CDNA5 async global->LDS path detection -------------------------------
#if defined(__HIP_DEVICE_COMPILE__) && __has_builtin(__builtin_amdgcn_global_load_async_to_lds_b128)
#define ASYNC_LDS 1
#else
#define ASYNC_LDS 0
#endif

#if defined(__HIP_DEVICE_COMPILE__) && __has_builtin(__builtin_amdgcn_s_wait_asynccnt)
#define WAIT_ASYNC(n) __builtin_amdgcn_s_wait_asynccnt(n)
#elif defined(__HIP_DEVICE_COMPILE__)
#define WAIT_ASYNC(n) asm volatile("s_wait_asynccnt " #n ::: "memory")
#else
#define WAIT_ASYNC(n)
#endif

typedef int v4i __attribute__((vector_size(16)));
typedef __attribute__((address_space(1))) v4i as1_v4i;   // global int4 (builtin param 0)
typedef __attribute__((address_space(3))) v4i as3_v4i;   // LDS int4    (builtin param 1)

// ---- Kernel 1: zero a float buffer ----------------------------------------
__global__ void msgagg_zero_kernel(float* __restrict__ a, int n) {
    int i = blockIdx.x * blockDim.x + threadIdx.x;
    if (i < n) a[i] = 0.0f;
}

// ---- Kernel 2: scatter-add sums and counts --------------------------------
__global__ __launch_bounds__(256) void msgagg_accum_kernel(
    const float* __restrict__ msgs, const int* __restrict__ seg,
    float* __restrict__ sums, float* __restrict__ counts, int n_msgs)
{
    const int lane = threadIdx.x & 31;
    const int wv   = threadIdx.x >> 5;              // wave index in block (8 waves)
    const int gw   = blockIdx.x * 8 + wv;           // global wave id
    const int nw   = gridDim.x * 8;                 // total waves
    const int npairs = n_msgs >> 1;                 // each iteration: 2 rows

#if ASYNC_LDS
    // 8 waves * 2 buffers * 128 floats (2 rows of 64) = 8 KB
    __shared__ __align__(16) float tile[8 * 2 * 2 * DM];
    float* buf = &tile[wv * 256];
    int p = 0;
    // Prologue: issue first async tile (512B: lane L loads 16B at offset L*16)
    if (gw < npairs) {
        const float* g = msgs + (size_t)gw * (2 * DM) + lane * 4;
        __builtin_amdgcn_global_load_async_to_lds_b128(
            (as1_v4i*)(unsigned long long)(const void*)g,
            (as3_v4i*)(buf + lane * 4), 0, 0);
    }
    for (int pr = gw; pr < npairs; pr += nw) {
        int nxt = pr + nw;
        if (nxt < npairs) {
            const float* g = msgs + (size_t)nxt * (2 * DM) + lane * 4;
            __builtin_amdgcn_global_load_async_to_lds_b128(
                (as1_v4i*)(unsigned long long)(const void*)g,
                (as3_v4i*)(buf + ((p ^ 1) * 128) + lane * 4), 0, 0);
            WAIT_ASYNC(1);   // oldest (current buffer) finished; next in flight
        } else {
            WAIT_ASYNC(0);
        }
        const float* cur = buf + p * 128;
        int2 ss = *(const int2*)(seg + (size_t)pr * 2);      // one b64: both seg ids
        float2 v0 = *(const float2*)(cur + lane * 2);        // row0 cols 2L,2L+1
        float2 v1 = *(const float2*)(cur + DM + lane * 2);   // row1 cols 2L,2L+1
        float* d0 = sums + (size_t)ss.x * DM + lane * 2;
        float* d1 = sums + (size_t)ss.y * DM + lane * 2;
        atomicAdd(d0,     v0.x);
        atomicAdd(d0 + 1, v0.y);
        atomicAdd(d1,     v1.x);
        atomicAdd(d1 + 1, v1.y);
        if (lane == 0) {
            atomicAdd(counts + ss.x, 1.0f);
            atomicAdd(counts + ss.y, 1.0f);
        }
        p ^= 1;
    }
#else
    // Fallback: direct vector loads (no async path available)
    for (int pr = gw; pr < npairs; pr += nw) {
        int m0 = pr * 2, m1 = m0 + 1;
        float2 v0 = *(const float2*)(msgs + (size_t)m0 * DM + lane * 2);
        float2 v1 = *(const float2*)(msgs + (size_t)m1 * DM + lane * 2);
        int2 ss = *(const int2*)(seg + (size_t)pr * 2);
        float* d0 = sums + (size_t)ss.x * DM + lane * 2;
        float* d1 = sums + (size_t)ss.y * DM + lane * 2;
        atomicAdd(d0,     v0.x);
        atomicAdd(d0 + 1, v0.y);
        atomicAdd(d1,     v1.x);
        atomicAdd(d1 + 1, v1.y);
        if (lane == 0) {
            atomicAdd(counts + ss.x, 1.0f);
            atomicAdd(counts + ss.y, 1.0f);
        }
    }
#endif
    // Odd-message tail (N_MSGS is even, but stay general)
    if ((n_msgs & 1) && gw == 0) {
        int m = n_msgs - 1;
        int s = seg[m];
        float2 v = *(const float2*)(msgs + (size_t)m * DM + lane * 2);
        float* d = sums + (size_t)s * DM + lane * 2;
        atomicAdd(d,     v.x);
        atomicAdd(d + 1, v.y);
        if (lane == 0) atomicAdd(counts + s, 1.0f);
    }
}

// ---- Kernel 3: divide by max(count, 1) ------------------------------------
__global__ void msgagg_div_kernel(float* __restrict__ out,
                                  const float* __restrict__ counts, int total) {
    int i = blockIdx.x * blockDim.x + threadIdx.x;
    if (i < total) {
        float c = counts[i >> 6];          // DM == 64
        c = c > 1.0f ? c : 1.0f;
        out[i] = out[i] / c;
    }
}

extern "C" void kernel_launch(void* const* d_in, const int* in_sizes, int n_in,
                              void* d_out, int out_size, void* d_ws, size_t ws_size,
                              hipStream_t stream) {
    (void)n_in; (void)ws_size;
    const float* msgs = (const float*)d_in[0];
    const int*   seg  = (const int*)d_in[1];
    const int n_msgs  = in_sizes[1];
    const int n_nodes = out_size / DM;

    float* out    = (float*)d_out;
    float* counts = (float*)d_ws;   // n_nodes floats of scratch

    // Zero accumulators (d_out and counts are poisoned by the harness)
    msgagg_zero_kernel<<<(out_size + 255) / 256, 256, 0, stream>>>(out, out_size);
    msgagg_zero_kernel<<<(n_nodes + 255) / 256, 256, 0, stream>>>(counts, n_nodes);

    // Scatter-accumulate: 4096 blocks * 8 waves = 32768 waves, grid-strided
    msgagg_accum_kernel<<<4096, 256, 0, stream>>>(msgs, seg, out, counts, n_msgs);

    // Mean
    msgagg_div_kernel<<<(out_size + 255) / 256, 256, 0, stream>>>(out, counts, out_size);
}